// PointNetFeaturePropagation_53334903881918
// MI455X (gfx1250) — compile-verified
//
#include <hip/hip_runtime.h>
#include <hip/hip_bf16.h>

typedef __attribute__((ext_vector_type(2))) float v2f;
typedef __attribute__((ext_vector_type(8))) float v8f;

#define B_    16
#define N_    4096
#define S_    1024
#define D1_   128
#define D2_   256
#define INCH_ 384
#define H1_   256
#define H2_   128
#define NPTS  (B_ * N_)   // 65536

// ---------------------------------------------------------------------------
// Fragment loader shared by A (row-major MxK) and B (B[k][n] = W[n][k], W
// row-major NxK) for V_WMMA_F32_16X16X4_F32.
// ISA layout (32-bit A 16x4): lanes 0-15 hold row=lane, v0=K0, v1=K1;
// lanes 16-31 hold row=lane-16, v0=K2, v1=K3. One b64 load per lane.
// ---------------------------------------------------------------------------
__device__ __forceinline__ v2f frag_ld(const float* __restrict__ base, int ld,
                                       int row0, int k, int lane) {
  const float* p = base + (size_t)(row0 + (lane & 15)) * (size_t)ld
                        + (size_t)(k + ((lane >> 4) << 1));
  return *reinterpret_cast<const v2f*>(p);
}

// ---------------------------------------------------------------------------
// Kernel 1: 3-NN search + gather + mean.  One block = 256 points of 1 batch.
// ---------------------------------------------------------------------------
__global__ void __launch_bounds__(256)
fp_knn_interp(const float* __restrict__ xyz1, const float* __restrict__ xyz2,
              const float* __restrict__ feat2, float* __restrict__ interp) {
  __shared__ float sx[S_], sy[S_], sz[S_];
  __shared__ int   sidx[256][3];

  const int b    = blockIdx.x >> 4;   // 16 point-tiles per batch
  const int tile = blockIdx.x & 15;
  const int tid  = threadIdx.x;

  const float* x2 = xyz2 + (size_t)b * S_ * 3;
  for (int s = tid; s < S_; s += 256) {
    sx[s] = x2[s * 3 + 0];
    sy[s] = x2[s * 3 + 1];
    sz[s] = x2[s * 3 + 2];
  }
  __syncthreads();

  const int n = tile * 256 + tid;                  // point within batch
  const float* p = xyz1 + ((size_t)b * N_ + n) * 3;
  const float px = p[0], py = p[1], pz = p[2];

  float d0 = 3.402823466e+38f, d1 = d0, d2 = d0;
  int   i0 = 0, i1 = 0, i2 = 0;
  for (int s = 0; s < S_; ++s) {                   // uniform loop: LDS broadcast
    const float dx = px - sx[s], dy = py - sy[s], dz = pz - sz[s];
    const float d = dx * dx + dy * dy + dz * dz;
    if (d < d2) {
      if (d < d1) {
        d2 = d1; i2 = i1;
        if (d < d0) { d1 = d0; i1 = i0; d0 = d; i0 = s; }
        else        { d1 = d;  i1 = s; }
      } else       { d2 = d;  i2 = s; }
    }
  }
  sidx[tid][0] = i0; sidx[tid][1] = i1; sidx[tid][2] = i2;
  __syncthreads();

  // Cooperative gather: thread == channel -> coalesced rows of 256 floats.
  const float* f2b = feat2 + (size_t)b * S_ * D2_;
  float* ob = interp + ((size_t)b * N_ + (size_t)tile * 256) * D2_;
  for (int pt = 0; pt < 256; ++pt) {
    const int j0 = sidx[pt][0], j1 = sidx[pt][1], j2 = sidx[pt][2];
    const float v = (f2b[(size_t)j0 * D2_ + tid] +
                     f2b[(size_t)j1 * D2_ + tid] +
                     f2b[(size_t)j2 * D2_ + tid]) * (1.0f / 3.0f);
    ob[(size_t)pt * D2_ + tid] = v;
  }
}

// ---------------------------------------------------------------------------
// Kernel 2: h[p][o] = sum_c [f1|interp][p][c] * W1[o][c] + b1[o]
// Block = 256 thr = 8 waves; wave tile 64(M points) x 64(N channels).
// Block tile 128 x 256 (full channel dim).
// ---------------------------------------------------------------------------
__global__ void __launch_bounds__(256)
fp_gemm1(const float* __restrict__ f1, const float* __restrict__ interp,
         const float* __restrict__ W1, const float* __restrict__ b1,
         float* __restrict__ h) {
  const int lane  = threadIdx.x & 31;
  const int wave  = threadIdx.x >> 5;
  const int mBase = blockIdx.x * 128 + (wave & 1) * 64;  // point rows
  const int nBase = (wave >> 1) * 64;                    // out channels

  const v8f vz = {0.f, 0.f, 0.f, 0.f, 0.f, 0.f, 0.f, 0.f};
  v8f acc[4][4];
#pragma unroll
  for (int i = 0; i < 4; ++i)
#pragma unroll
    for (int j = 0; j < 4; ++j) acc[i][j] = vz;

  // K-range 1: A from features1 (ld=128)
  for (int k = 0; k < D1_; k += 4) {
    v2f a[4], bm[4];
#pragma unroll
    for (int i = 0; i < 4; ++i) a[i]  = frag_ld(f1, D1_, mBase + i * 16, k, lane);
#pragma unroll
    for (int j = 0; j < 4; ++j) bm[j] = frag_ld(W1, INCH_, nBase + j * 16, k, lane);
#pragma unroll
    for (int i = 0; i < 4; ++i)
#pragma unroll
      for (int j = 0; j < 4; ++j)
        acc[i][j] = __builtin_amdgcn_wmma_f32_16x16x4_f32(
            false, a[i], false, bm[j], (short)0, acc[i][j], false, false);
  }
  // K-range 2: A from interp (ld=256), B from W1 columns 128..383
  for (int k = 0; k < D2_; k += 4) {
    v2f a[4], bm[4];
#pragma unroll
    for (int i = 0; i < 4; ++i) a[i]  = frag_ld(interp, D2_, mBase + i * 16, k, lane);
#pragma unroll
    for (int j = 0; j < 4; ++j) bm[j] = frag_ld(W1, INCH_, nBase + j * 16, D1_ + k, lane);
#pragma unroll
    for (int i = 0; i < 4; ++i)
#pragma unroll
      for (int j = 0; j < 4; ++j)
        acc[i][j] = __builtin_amdgcn_wmma_f32_16x16x4_f32(
            false, a[i], false, bm[j], (short)0, acc[i][j], false, false);
  }

  // Epilogue: D vgpr r -> row = mBase+i*16 + (lane>=16 ? 8 : 0) + r, col = lane&15
  const int cl = lane & 15;
  const int rh = (lane >> 4) << 3;
#pragma unroll
  for (int j = 0; j < 4; ++j) {
    const int col  = nBase + j * 16 + cl;
    const float bv = b1[col];
#pragma unroll
    for (int i = 0; i < 4; ++i) {
#pragma unroll
      for (int r = 0; r < 8; ++r) {
        const int row = mBase + i * 16 + rh + r;
        h[(size_t)row * H1_ + col] = acc[i][j][r] + bv;
      }
    }
  }
}

// ---------------------------------------------------------------------------
// Kernel 3: out[b][o][n] = sum_c W2[o][c]*h[b*N+n][c] + b2[o]
// Roles swapped (A = W2, B = h^T) so D is channel-major == output layout and
// stores are coalesced along the point dimension.
// Block = 8 waves: 2 along M (covers all 128 channels) x 4 along N (256 pts).
// ---------------------------------------------------------------------------
__global__ void __launch_bounds__(256)
fp_gemm2(const float* __restrict__ h, const float* __restrict__ W2,
         const float* __restrict__ b2, float* __restrict__ out) {
  const int lane  = threadIdx.x & 31;
  const int wave  = threadIdx.x >> 5;
  const int cBase = (wave & 1) * 64;                      // channel rows
  const int pBase = blockIdx.x * 256 + (wave >> 1) * 64;  // global points

  const v8f vz = {0.f, 0.f, 0.f, 0.f, 0.f, 0.f, 0.f, 0.f};
  v8f acc[4][4];
#pragma unroll
  for (int i = 0; i < 4; ++i)
#pragma unroll
    for (int j = 0; j < 4; ++j) acc[i][j] = vz;

  for (int k = 0; k < H1_; k += 4) {
    v2f a[4], bm[4];
#pragma unroll
    for (int i = 0; i < 4; ++i) a[i]  = frag_ld(W2, H1_, cBase + i * 16, k, lane);
#pragma unroll
    for (int j = 0; j < 4; ++j) bm[j] = frag_ld(h, H1_, pBase + j * 16, k, lane);
#pragma unroll
    for (int i = 0; i < 4; ++i)
#pragma unroll
      for (int j = 0; j < 4; ++j)
        acc[i][j] = __builtin_amdgcn_wmma_f32_16x16x4_f32(
            false, a[i], false, bm[j], (short)0, acc[i][j], false, false);
  }

  const int cl = lane & 15;
  const int rh = (lane >> 4) << 3;
#pragma unroll
  for (int j = 0; j < 4; ++j) {
    const int p  = pBase + j * 16 + cl;  // global point
    const int bb = p >> 12;              // / N_
    const int nn = p & (N_ - 1);
#pragma unroll
    for (int i = 0; i < 4; ++i) {
#pragma unroll
      for (int r = 0; r < 8; ++r) {
        const int ch = cBase + i * 16 + rh + r;
        out[((size_t)bb * H2_ + ch) * N_ + nn] = acc[i][j][r] + b2[ch];
      }
    }
  }
}

// ---------------------------------------------------------------------------
extern "C" void kernel_launch(void* const* d_in, const int* in_sizes, int n_in,
                              void* d_out, int out_size, void* d_ws, size_t ws_size,
                              hipStream_t stream) {
  const float* xyz1 = (const float*)d_in[0];
  const float* xyz2 = (const float*)d_in[1];
  const float* f1   = (const float*)d_in[2];
  const float* f2   = (const float*)d_in[3];
  const float* W1   = (const float*)d_in[4];
  const float* b1   = (const float*)d_in[5];
  const float* W2   = (const float*)d_in[6];
  const float* b2   = (const float*)d_in[7];
  float* out = (float*)d_out;

  float* interp = (float*)d_ws;                       // [65536, 256]  64 MB
  float* h      = interp + (size_t)NPTS * D2_;        // [65536, 256]  64 MB

  fp_knn_interp<<<B_ * (N_ / 256), 256, 0, stream>>>(xyz1, xyz2, f2, interp);
  fp_gemm1<<<NPTS / 128, 256, 0, stream>>>(f1, interp, W1, b1, h);
  fp_gemm2<<<NPTS / 256, 256, 0, stream>>>(h, W2, b2, out);
}